// HaarWaveletGPU_18502719111405
// MI455X (gfx1250) — compile-verified
//
#include <hip/hip_runtime.h>
#include <stdint.h>

typedef float __attribute__((ext_vector_type(4))) f4;

#define WVS 8          // waves per block
#define STAGES 4       // row-pairs per wave (software-pipelined)

constexpr int W      = 512;
constexpr int H      = 512;
constexpr int OW     = 256;
constexpr int OPLANE = OW * OW;      // 65536 floats per output plane

// Issue one row-pair (4KB) as 8 contiguous async b128 copies, global -> LDS.
// INST_OFFSET is applied to BOTH the global and LDS address (ISA 15.18),
// so one lds-addr VGPR and one 64-bit vaddr pair cover the whole stage.
#define ISSUE_STAGE(ldsaddr, gaddr)                                                                   \
  do {                                                                                                \
    unsigned _l = (ldsaddr);                                                                          \
    uint64_t _g = (gaddr);                                                                            \
    asm volatile("global_load_async_to_lds_b128 %0, %1, off"             :: "v"(_l), "v"(_g) : "memory"); \
    asm volatile("global_load_async_to_lds_b128 %0, %1, off offset:512"  :: "v"(_l), "v"(_g) : "memory"); \
    asm volatile("global_load_async_to_lds_b128 %0, %1, off offset:1024" :: "v"(_l), "v"(_g) : "memory"); \
    asm volatile("global_load_async_to_lds_b128 %0, %1, off offset:1536" :: "v"(_l), "v"(_g) : "memory"); \
    asm volatile("global_load_async_to_lds_b128 %0, %1, off offset:2048" :: "v"(_l), "v"(_g) : "memory"); \
    asm volatile("global_load_async_to_lds_b128 %0, %1, off offset:2560" :: "v"(_l), "v"(_g) : "memory"); \
    asm volatile("global_load_async_to_lds_b128 %0, %1, off offset:3072" :: "v"(_l), "v"(_g) : "memory"); \
    asm volatile("global_load_async_to_lds_b128 %0, %1, off offset:3584" :: "v"(_l), "v"(_g) : "memory"); \
  } while (0)

__global__ __launch_bounds__(256) void haar_dwt_kernel(const float* __restrict__ x,
                                                       float* __restrict__ out) {
  // 8 waves * double-buffer * 1024 floats (one row-pair) = 64 KB
  __shared__ __align__(16) float smem[WVS * 2 * 1024];

  const int tid   = threadIdx.x;
  const int lane  = tid & 31;
  const int wv    = tid >> 5;
  const int plane = blockIdx.x;                       // 0..511  (= b*64 + c)
  const int chunk = blockIdx.y;                       // 0..7
  const int rp0   = chunk * (WVS * STAGES) + wv * STAGES;  // first row-pair of this wave

  const float* gplane = x + (size_t)plane * (size_t)(H * W);

  // Per-stage per-lane global base addresses (lane covers 16B chunk L).
  uint64_t ga[STAGES];
#pragma unroll
  for (int s = 0; s < STAGES; ++s)
    ga[s] = (uint64_t)(uintptr_t)(gplane + (size_t)(rp0 + s) * (2 * W) + lane * 4);

  // Raw LDS byte addresses (low 32 bits of the generic pointer) for the two buffers.
  const unsigned lds0 = (unsigned)(uintptr_t)&smem[wv * 2048 + lane * 4];
  const unsigned ldsbuf[2] = { lds0, lds0 + 4096u };

  // ---- software pipeline: prefetch depth 1, ASYNCcnt-gated ----
  ISSUE_STAGE(ldsbuf[0], ga[0]);

#pragma unroll
  for (int s = 0; s < STAGES; ++s) {
    if (s + 1 < STAGES) {
      // Re-filling buffer (s+1)&1: make sure our earlier ds reads of it retired (WAR).
      asm volatile("s_wait_dscnt 0x0" ::: "memory");
      ISSUE_STAGE(ldsbuf[(s + 1) & 1], ga[s + 1]);
      asm volatile("s_wait_asynccnt 0x8" ::: "memory");   // stage s complete, s+1 in flight
    } else {
      asm volatile("s_wait_asynccnt 0x0" ::: "memory");   // drain
    }

    const float* buf = &smem[wv * 2048 + (s & 1) * 1024]; // row0[512] | row1[512]
    const f4* r0 = (const f4*)(buf + lane * 16);
    const f4* r1 = (const f4*)(buf + 512 + lane * 16);

    float ll[8], lh[8], hl[8], hh[8];
#pragma unroll
    for (int i = 0; i < 4; ++i) {
      f4 a = r0[i];           // row 2h : x00 x01 x00 x01
      f4 b = r1[i];           // row 2h+1: x10 x11 x10 x11
      // pixel 2i
      float m0 = a.x + a.y, m1 = a.x - a.y;
      float n0 = b.x + b.y, n1 = b.x - b.y;
      ll[2 * i]     =  0.5f * (m0 + n0);
      lh[2 * i]     =  0.5f * (n0 - m0);
      hl[2 * i]     = -0.5f * (m1 + n1);
      hh[2 * i]     =  0.5f * (m1 - n1);
      // pixel 2i+1
      m0 = a.z + a.w; m1 = a.z - a.w;
      n0 = b.z + b.w; n1 = b.z - b.w;
      ll[2 * i + 1] =  0.5f * (m0 + n0);
      lh[2 * i + 1] =  0.5f * (n0 - m0);
      hl[2 * i + 1] = -0.5f * (m1 + n1);
      hh[2 * i + 1] =  0.5f * (m1 - n1);
    }

    const int    orow = rp0 + s;
    const size_t ob   = (size_t)(plane * 4) * OPLANE + (size_t)orow * OW + (lane << 3);
    float* p0 = out + ob;                 // LL plane (4c+0)
    float* p1 = p0 + OPLANE;              // LH plane (4c+1)
    float* p2 = p1 + OPLANE;              // HL plane (4c+2)
    float* p3 = p2 + OPLANE;              // HH plane (4c+3)

    f4 v;
    v = (f4){ll[0], ll[1], ll[2], ll[3]}; __builtin_nontemporal_store(v, (f4*)p0);
    v = (f4){ll[4], ll[5], ll[6], ll[7]}; __builtin_nontemporal_store(v, (f4*)(p0 + 4));
    v = (f4){lh[0], lh[1], lh[2], lh[3]}; __builtin_nontemporal_store(v, (f4*)p1);
    v = (f4){lh[4], lh[5], lh[6], lh[7]}; __builtin_nontemporal_store(v, (f4*)(p1 + 4));
    v = (f4){hl[0], hl[1], hl[2], hl[3]}; __builtin_nontemporal_store(v, (f4*)p2);
    v = (f4){hl[4], hl[5], hl[6], hl[7]}; __builtin_nontemporal_store(v, (f4*)(p2 + 4));
    v = (f4){hh[0], hh[1], hh[2], hh[3]}; __builtin_nontemporal_store(v, (f4*)p3);
    v = (f4){hh[4], hh[5], hh[6], hh[7]}; __builtin_nontemporal_store(v, (f4*)(p3 + 4));
  }
}

extern "C" void kernel_launch(void* const* d_in, const int* in_sizes, int n_in,
                              void* d_out, int out_size, void* d_ws, size_t ws_size,
                              hipStream_t stream) {
  (void)in_sizes; (void)n_in; (void)out_size; (void)d_ws; (void)ws_size;
  const float* x = (const float*)d_in[0];
  float* out = (float*)d_out;

  // 512 planes (B*C) x 8 row-chunks; 256 threads = 8 waves of 32 (wave32).
  dim3 grid(512, 8, 1);
  dim3 block(256, 1, 1);
  hipLaunchKernelGGL(haar_dwt_kernel, grid, block, 0, stream, x, out);
}